// DSSIMLoss_26156350833263
// MI455X (gfx1250) — compile-verified
//
#include <hip/hip_runtime.h>

typedef __attribute__((ext_vector_type(2))) float v2f;
typedef __attribute__((ext_vector_type(8))) float v8f;

#define HH 2160
#define WW 3840
#define CC 3
#define TS 16
#define HALO 28            // 16 + 2*5 rounded up to multiple of 4 for K-chunking
#define TXN (WW / TS)      // 240
#define TYN (HH / TS)      // 135
#define NUNITS (TXN * TYN * CC)  // 97200
#define WPB 4              // waves per block (wave32)
#define NPIX (float)(HH * WW * CC)

// ---- compile-time banded Gaussian weight table -----------------------------
// Normalized 11-tap Gaussian, sigma=1.5 (matches exp(-d^2/4.5)/sum in fp32).
// Band matrix A[j][m] = G[j-m] for j-m in [0,10], used as the WMMA A operand
// for BOTH separable passes. Stored as [j/2][m][2] so each lane grabs its two
// K-slot weights with one 8-byte load (j0 = 4*kc + 2*half is always even).
struct WTab {
  float v[HALO / 2][16][2];
};
static constexpr WTab makeW() {
  WTab t{};
  constexpr float G[11] = {0.00102838f, 0.00759879f, 0.03600080f, 0.10936051f,
                           0.21300565f, 0.26601192f, 0.21300565f, 0.10936051f,
                           0.03600080f, 0.00759879f, 0.00102838f};
  for (int p = 0; p < HALO / 2; ++p)
    for (int m = 0; m < 16; ++m)
      for (int tt = 0; tt < 2; ++tt) {
        const int d = (2 * p + tt) - m;
        t.v[p][m][tt] = (d >= 0 && d <= 10) ? G[d] : 0.0f;
      }
  return t;
}
__device__ __constant__ WTab WT = makeW();

static __device__ __forceinline__ v8f wmma4(v2f a, v2f b, v8f c) {
  // D = A(16x4 f32) * B(4x16 f32) + C(16x16 f32) on the CDNA5 matrix pipe
  return __builtin_amdgcn_wmma_f32_16x16x4_f32(false, a, false, b, (short)0, c,
                                               false, false);
}

__global__ __launch_bounds__(WPB * 32) void DSSIM_zero(float* acc) {
  if (threadIdx.x == 0) acc[0] = 0.0f;
}

__global__ __launch_bounds__(WPB * 32) void DSSIM_main(
    const float* __restrict__ X, const float* __restrict__ Y,
    float* __restrict__ acc) {
  // wave-private LDS: input halos + vertical-blur temporaries (5 streams)
  __shared__ float xs_[WPB][HALO * HALO];
  __shared__ float ys_[WPB][HALO * HALO];
  __shared__ float vt_[WPB][5][TS * HALO];

  const int wave = threadIdx.x >> 5;
  const int lane = threadIdx.x & 31;
  const int lm   = lane & 15;   // M (A/D) or N (B/D) sub-index
  const int half = lane >> 4;   // K-half selector

  // unit -> (channel, tileY, tileX); adjacent waves take adjacent columns
  const int unit = blockIdx.x * WPB + wave;
  const int ux   = unit % TXN;
  const int rest = unit / TXN;
  const int uy   = rest % TYN;
  const int ch   = rest / TYN;

  const int h0 = uy * TS - 5;
  const int w0 = ux * TS - 5;

  float* xs = xs_[wave];
  float* ys = ys_[wave];

  // ---- phase 1: load 28x28 halos (zero padded, matching SAME zero-pad) ----
  // One column per lane (28 active), rows swept sequentially: pure strided
  // addressing, coalesced across lanes, row-major order for L2 locality.
  if (lane < HALO) {
    const int w = w0 + lane;
    const bool wok = (unsigned)w < (unsigned)WW;
#pragma unroll
    for (int r = 0; r < HALO; ++r) {
      const int h = h0 + r;
      float xv = 0.0f, yv = 0.0f;
      if (wok && (unsigned)h < (unsigned)HH) {
        const long idx = ((long)h * WW + w) * CC + ch;
        xv = X[idx];
        yv = Y[idx];
      }
      xs[r * HALO + lane] = xv;
      ys[r * HALO + lane] = yv;
    }
  }

  // ---- per-lane A-operand weights: one b64 constant load per K-chunk ----
  // A layout (32-bit A 16x4): lane<16 -> K=t ; lane>=16 -> K=2+t
  v2f wA[7];
#pragma unroll
  for (int kc = 0; kc < 7; ++kc) {
    const int j0 = 4 * kc + 2 * half;  // even
    const float* p = &WT.v[j0 >> 1][lm][0];
    wA[kc][0] = p[0];
    wA[kc][1] = p[1];
  }

  __builtin_amdgcn_wave_barrier();

  // ---- phase 2: vertical blur via WMMA, two 16-col groups, 5 streams ----
#pragma unroll
  for (int cg = 0; cg < 2; ++cg) {
    const int co = cg * 12;  // col groups cover 0..15 and 12..27
    v8f D0 = {}, D1 = {}, D2 = {}, D3 = {}, D4 = {};
#pragma unroll
    for (int kc = 0; kc < 7; ++kc) {
      const int j0 = 4 * kc + 2 * half;      // input row for this lane's K slot
      const int b0 = j0 * HALO + co + lm;    // LDS index (row j0, col co+lm)
      v2f bx, by;
      bx[0] = xs[b0]; bx[1] = xs[b0 + HALO];
      by[0] = ys[b0]; by[1] = ys[b0 + HALO];
      const v2f a = wA[kc];
      D0 = wmma4(a, bx, D0);            // E[x]
      D1 = wmma4(a, by, D1);            // E[y]
      D2 = wmma4(a, bx * bx, D2);       // E[x^2]
      D3 = wmma4(a, by * by, D3);       // E[y^2]
      D4 = wmma4(a, bx * by, D4);       // E[x*y]
    }
    // D layout: VGPR r, lanes<16 -> M=r, lanes>=16 -> M=r+8 ; N = lm
#pragma unroll
    for (int r = 0; r < 8; ++r) {
      const int o = (r + 8 * half) * HALO + co + lm;
      vt_[wave][0][o] = D0[r];
      vt_[wave][1][o] = D1[r];
      vt_[wave][2][o] = D2[r];
      vt_[wave][3][o] = D3[r];
      vt_[wave][4][o] = D4[r];
    }
  }

  __builtin_amdgcn_wave_barrier();

  // ---- phase 3: horizontal blur, computed transposed: Out^T = G * Vtmp^T ----
  v8f O0 = {}, O1 = {}, O2 = {}, O3 = {}, O4 = {};
#pragma unroll
  for (int kc = 0; kc < 7; ++kc) {
    const int j0 = 4 * kc + 2 * half;
    const int b  = lm * HALO + j0;  // transposed read: B'[k][m] = vtmp[m][j]
    const v2f a = wA[kc];
    v2f B;
    B[0] = vt_[wave][0][b]; B[1] = vt_[wave][0][b + 1];
    O0 = wmma4(a, B, O0);
    B[0] = vt_[wave][1][b]; B[1] = vt_[wave][1][b + 1];
    O1 = wmma4(a, B, O1);
    B[0] = vt_[wave][2][b]; B[1] = vt_[wave][2][b + 1];
    O2 = wmma4(a, B, O2);
    B[0] = vt_[wave][3][b]; B[1] = vt_[wave][3][b + 1];
    O3 = wmma4(a, B, O3);
    B[0] = vt_[wave][4][b]; B[1] = vt_[wave][4][b + 1];
    O4 = wmma4(a, B, O4);
  }

  // ---- phase 4: SSIM map (register-local; orientation irrelevant for mean) --
  const float C1v = 0.01f * 0.01f;
  const float C2v = 0.03f * 0.03f;
  float sum = 0.0f;
#pragma unroll
  for (int r = 0; r < 8; ++r) {
    const float mux = O0[r], muy = O1[r];
    const float exx = O2[r], eyy = O3[r], exy = O4[r];
    const float mxx = mux * mux, myy = muy * muy, mxy = mux * muy;
    const float sxx = exx - mxx, syy = eyy - myy, sxy = exy - mxy;
    const float num = (2.0f * mxy + C1v) * (2.0f * sxy + C2v);
    const float den = (mxx + myy + C1v) * (sxx + syy + C2v);
    sum += num / den;
  }
  // wave32 reduction, one atomic per wave
#pragma unroll
  for (int off = 16; off > 0; off >>= 1) sum += __shfl_xor(sum, off, 32);
  if (lane == 0) unsafeAtomicAdd(acc, sum);
}

__global__ __launch_bounds__(32) void DSSIM_final(const float* __restrict__ acc,
                                                  float* __restrict__ out) {
  if (threadIdx.x == 0) out[0] = 1.0f - acc[0] * (1.0f / NPIX);
}

extern "C" void kernel_launch(void* const* d_in, const int* in_sizes, int n_in,
                              void* d_out, int out_size, void* d_ws,
                              size_t ws_size, hipStream_t stream) {
  const float* X = (const float*)d_in[0];
  const float* Y = (const float*)d_in[1];
  float* out = (float*)d_out;
  float* acc = (float*)d_ws;  // 4-byte fp32 accumulator in scratch

  DSSIM_zero<<<1, WPB * 32, 0, stream>>>(acc);
  DSSIM_main<<<NUNITS / WPB, WPB * 32, 0, stream>>>(X, Y, acc);
  DSSIM_final<<<1, 32, 0, stream>>>(acc, out);
}